// HSICLoss_24584392802988
// MI455X (gfx1250) — compile-verified
//
#include <hip/hip_runtime.h>

typedef __attribute__((ext_vector_type(2))) float v2f;
typedef __attribute__((ext_vector_type(8))) float v8f;

#define NROW 4096
#define DIM  512
#define NTOT 8192
#define REGN (NROW / 32)   // 128 regions of 32 rows

// workspace layout (float offsets)
#define WS_S1   0          // sum of all row sum-squares (8192 rows)
#define WS_P    1          // <KW,KG> accumulator
#define WS_NI   2          // 5 floats: -1/(bw0*2^i)
#define WS_SQ   8          // 8192: sqW[4096] then sqG[4096]
#define WS_CS   8200       // 512 column sums of total
#define WS_RW   8712       // 4096 row sums of KW
#define WS_RG   12808      // 4096 row sums of KG
#define WS_END  16904

__global__ void zero_ws(float* __restrict__ ws) {
  int i = blockIdx.x * blockDim.x + threadIdx.x;
  if (i < WS_END) ws[i] = 0.f;
}

// one wave per row: row sum-of-squares + global S1
__global__ void row_stats(const float* __restrict__ W, const float* __restrict__ G,
                          float* __restrict__ sq, float* __restrict__ S1) {
  int row  = (blockIdx.x * blockDim.x + threadIdx.x) >> 5;
  int lane = threadIdx.x & 31;
  if (row >= NTOT) return;
  const float4* p = (const float4*)((row < NROW) ? (W + (size_t)row * DIM)
                                                 : (G + (size_t)(row - NROW) * DIM));
  float s = 0.f;
  #pragma unroll
  for (int k = 0; k < DIM / 128; ++k) {          // 512 floats = 128 float4, 4 per lane
    float4 x = p[lane + k * 32];
    s += x.x * x.x + x.y * x.y + x.z * x.z + x.w * x.w;
  }
  #pragma unroll
  for (int m = 16; m >= 1; m >>= 1) s += __shfl_xor(s, m, 32);
  if (lane == 0) { sq[row] = s; atomicAdd(S1, s); }
}

// column sums of total = [W;G]; 64 blocks = 2 column halves x 32 row chunks of 256
__global__ void col_stats(const float* __restrict__ W, const float* __restrict__ G,
                          float* __restrict__ cs) {
  int c  = (blockIdx.x & 1) * 256 + threadIdx.x;
  int r0 = (blockIdx.x >> 1) * 256;
  float s = 0.f;
  for (int r = r0; r < r0 + 256; ++r) {
    const float* src = (r < NROW) ? (W + (size_t)r * DIM) : (G + (size_t)(r - NROW) * DIM);
    s += src[c];
  }
  atomicAdd(&cs[c], s);
}

// bandwidth: sum(d2) = 2*N*S1 - 2*||colsum||^2 ; bw0 = sum/(N^2-N)/4
__global__ void finalize_bw(float* __restrict__ ws) {
  __shared__ float red[512];
  int t = threadIdx.x;
  float c = ws[WS_CS + t];
  red[t] = c * c;
  __syncthreads();
  for (int s = 256; s > 0; s >>= 1) { if (t < s) red[t] += red[t + s]; __syncthreads(); }
  if (t == 0) {
    double Nf = (double)NTOT;
    double sumd2 = 2.0 * Nf * (double)ws[WS_S1] - 2.0 * (double)red[0];
    double bw0 = sumd2 / (Nf * Nf - Nf) / 4.0;   // / KERNEL_MUL^(KERNEL_NUM//2)
    #pragma unroll
    for (int i = 0; i < 5; ++i) ws[WS_NI + i] = (float)(-1.0 / (bw0 * (double)(1 << i)));
  }
}

// Main: per-wave 32x32 region of the 4096x4096 tile space, upper triangle only.
// 2x2 f32 WMMA tiles per matrix (W and G), K=512 in steps of 4.
__global__ __launch_bounds__(256)
void hsic_main(const float* __restrict__ W, const float* __restrict__ G,
               float* __restrict__ ws) {
  const float* sqW = ws + WS_SQ;
  const float* sqG = ws + WS_SQ + NROW;
  float* rW = ws + WS_RW;
  float* rG = ws + WS_RG;

  const int lane = threadIdx.x & 31;
  const int wid  = (int)((blockIdx.x * blockDim.x + threadIdx.x) >> 5);
  const int Ir = wid / REGN, Jr = wid % REGN;     // wave-uniform
  if (Ir >= REGN || Jr < Ir) return;              // triangle (symmetric kernels)

  const float ni0 = ws[WS_NI + 0], ni1 = ws[WS_NI + 1], ni2 = ws[WS_NI + 2],
              ni3 = ws[WS_NI + 3], ni4 = ws[WS_NI + 4];

  const int I0 = Ir * 32, J0 = Jr * 32;
  const int rlo  = lane & 15;          // row within 16-block / column N
  const int koff = (lane >> 4) * 2;    // k-pair select (A and B layouts coincide for X*X^T)

  const float* aW0 = W + (size_t)(I0 + rlo) * DIM + koff;
  const float* aW1 = aW0 + 16 * DIM;
  const float* bW0 = W + (size_t)(J0 + rlo) * DIM + koff;
  const float* bW1 = bW0 + 16 * DIM;
  const float* aG0 = G + (size_t)(I0 + rlo) * DIM + koff;
  const float* aG1 = aG0 + 16 * DIM;
  const float* bG0 = G + (size_t)(J0 + rlo) * DIM + koff;
  const float* bG1 = bG0 + 16 * DIM;

  v8f cw00 = {}, cw01 = {}, cw10 = {}, cw11 = {};
  v8f cg00 = {}, cg01 = {}, cg10 = {}, cg11 = {};

  #pragma unroll 4
  for (int k = 0; k < DIM; k += 4) {
    v2f a0 = *(const v2f*)(aW0 + k);
    v2f a1 = *(const v2f*)(aW1 + k);
    v2f b0 = *(const v2f*)(bW0 + k);
    v2f b1 = *(const v2f*)(bW1 + k);
    cw00 = __builtin_amdgcn_wmma_f32_16x16x4_f32(false, a0, false, b0, (short)0, cw00, false, false);
    cw01 = __builtin_amdgcn_wmma_f32_16x16x4_f32(false, a0, false, b1, (short)0, cw01, false, false);
    cw10 = __builtin_amdgcn_wmma_f32_16x16x4_f32(false, a1, false, b0, (short)0, cw10, false, false);
    cw11 = __builtin_amdgcn_wmma_f32_16x16x4_f32(false, a1, false, b1, (short)0, cw11, false, false);
    v2f e0 = *(const v2f*)(aG0 + k);
    v2f e1 = *(const v2f*)(aG1 + k);
    v2f f0 = *(const v2f*)(bG0 + k);
    v2f f1 = *(const v2f*)(bG1 + k);
    cg00 = __builtin_amdgcn_wmma_f32_16x16x4_f32(false, e0, false, f0, (short)0, cg00, false, false);
    cg01 = __builtin_amdgcn_wmma_f32_16x16x4_f32(false, e0, false, f1, (short)0, cg01, false, false);
    cg10 = __builtin_amdgcn_wmma_f32_16x16x4_f32(false, e1, false, f0, (short)0, cg10, false, false);
    cg11 = __builtin_amdgcn_wmma_f32_16x16x4_f32(false, e1, false, f1, (short)0, cg11, false, false);
  }

  // epilogue: d2 -> 5-bandwidth gaussian kernels -> reductions
  const int half8 = (lane >> 4) * 8;   // C layout: VGPR v holds M = v + half8, N = lane&15
  float sqiW[2][8], sqiG[2][8];
  #pragma unroll
  for (int r = 0; r < 2; ++r)
    #pragma unroll
    for (int v = 0; v < 8; ++v) {
      sqiW[r][v] = sqW[I0 + r * 16 + half8 + v];
      sqiG[r][v] = sqG[I0 + r * 16 + half8 + v];
    }

  float pAcc = 0.f;
  float rAW[2][8] = {}, rAG[2][8] = {};
  float cAW[2] = {0.f, 0.f}, cAG[2] = {0.f, 0.f};

#define PROC_TILE(r, c, CW, CG) do {                                          \
    float sqjW = sqW[J0 + (c) * 16 + rlo];                                    \
    float sqjG = sqG[J0 + (c) * 16 + rlo];                                    \
    _Pragma("unroll")                                                         \
    for (int v = 0; v < 8; ++v) {                                             \
      float d2w = fmaxf(sqiW[r][v] + sqjW - 2.f * CW[v], 0.f);                \
      float d2g = fmaxf(sqiG[r][v] + sqjG - 2.f * CG[v], 0.f);                \
      float kw = __expf(d2w * ni0) + __expf(d2w * ni1) + __expf(d2w * ni2) +  \
                 __expf(d2w * ni3) + __expf(d2w * ni4);                       \
      float kg = __expf(d2g * ni0) + __expf(d2g * ni1) + __expf(d2g * ni2) +  \
                 __expf(d2g * ni3) + __expf(d2g * ni4);                       \
      pAcc += kw * kg;                                                        \
      rAW[r][v] += kw; rAG[r][v] += kg;                                       \
      cAW[c] += kw;    cAG[c] += kg;                                          \
    }                                                                         \
  } while (0)

  PROC_TILE(0, 0, cw00, cg00);
  PROC_TILE(0, 1, cw01, cg01);
  PROC_TILE(1, 0, cw10, cg10);
  PROC_TILE(1, 1, cw11, cg11);
#undef PROC_TILE

  // <KW,KG>: full-wave reduce; off-diagonal regions count twice by symmetry
  #pragma unroll
  for (int m = 16; m >= 1; m >>= 1) pAcc += __shfl_xor(pAcc, m, 32);
  if (lane == 0) atomicAdd(ws + WS_P, ((Ir == Jr) ? 1.f : 2.f) * pAcc);

  // row sums: reduce across the 16 lanes of each half (columns of the region)
  #pragma unroll
  for (int r = 0; r < 2; ++r)
    #pragma unroll
    for (int v = 0; v < 8; ++v) {
      float sw = rAW[r][v], sg = rAG[r][v];
      #pragma unroll
      for (int m = 8; m >= 1; m >>= 1) {
        sw += __shfl_xor(sw, m, 32);
        sg += __shfl_xor(sg, m, 32);
      }
      if (rlo == 0) {
        atomicAdd(&rW[I0 + r * 16 + half8 + v], sw);
        atomicAdd(&rG[I0 + r * 16 + half8 + v], sg);
      }
    }

  // column sums mirror into rows J0.. (the skipped transpose region)
  if (Ir != Jr) {
    #pragma unroll
    for (int c = 0; c < 2; ++c) {
      float sw = cAW[c] + __shfl_xor(cAW[c], 16, 32);
      float sg = cAG[c] + __shfl_xor(cAG[c], 16, 32);
      if (lane < 16) {
        atomicAdd(&rW[J0 + c * 16 + lane], sw);
        atomicAdd(&rG[J0 + c * 16 + lane], sg);
      }
    }
  }
}

// loss = -( P - (2/n) rW.rG + sW*sG/n^2 ) / (n-1)^2
__global__ void finalize_loss(const float* __restrict__ ws, float* __restrict__ out) {
  __shared__ float sd[256], sa[256], sb[256];
  int t = threadIdx.x;
  float dot = 0.f, sw = 0.f, sg = 0.f;
  for (int i = t; i < NROW; i += 256) {
    float a = ws[WS_RW + i], b = ws[WS_RG + i];
    dot += a * b; sw += a; sg += b;
  }
  sd[t] = dot; sa[t] = sw; sb[t] = sg;
  __syncthreads();
  for (int s = 128; s > 0; s >>= 1) {
    if (t < s) { sd[t] += sd[t + s]; sa[t] += sa[t + s]; sb[t] += sb[t + s]; }
    __syncthreads();
  }
  if (t == 0) {
    double n  = (double)NROW;
    double tr = (double)ws[WS_P] - (2.0 / n) * (double)sd[0]
              + ((double)sa[0] * (double)sb[0]) / (n * n);
    out[0] = (float)(-tr / ((n - 1.0) * (n - 1.0)));
  }
}

extern "C" void kernel_launch(void* const* d_in, const int* in_sizes, int n_in,
                              void* d_out, int out_size, void* d_ws, size_t ws_size,
                              hipStream_t stream) {
  const float* W = (const float*)d_in[0];
  const float* G = (const float*)d_in[1];
  float* ws  = (float*)d_ws;
  float* out = (float*)d_out;

  zero_ws<<<(WS_END + 255) / 256, 256, 0, stream>>>(ws);
  row_stats<<<(NTOT * 32) / 256, 256, 0, stream>>>(W, G, ws + WS_SQ, ws + WS_S1);
  col_stats<<<64, 256, 0, stream>>>(W, G, ws + WS_CS);
  finalize_bw<<<1, 512, 0, stream>>>(ws);
  hsic_main<<<(REGN * REGN) / 8, 256, 0, stream>>>(W, G, ws);  // 8 waves/block
  finalize_loss<<<1, 256, 0, stream>>>(ws, out);
}